// SpatiallyAdaptiveTVNet2D_91027536872131
// MI455X (gfx1250) — compile-verified
//
#include <hip/hip_runtime.h>
#include <math.h>

// ---------------------------------------------------------------------------
// PDHG TV-regularized MRI recon on MI455X (gfx1250).
// 2D centered ortho FFTs as DFT matmuls on V_WMMA_F32_16X16X4_F32.
// Row-pass tile staging via the Tensor Data Mover (TENSOR_LOAD_TO_LDS /
// TENSOR_STORE_FROM_LDS), synchronized with s_wait_tensorcnt.
// Centered-FFT (-1)^(r+c) signs are folded into the twiddle table:
//   out[k] = (-1)^k sum_n (-1)^n x[n] w^{nk},  k=32k1+k2, n=n1+16n2
//   => (-1)^n = (-1)^{n1}, (-1)^k = (-1)^{k2}, both absorbed by TW[n1][k2].
// ---------------------------------------------------------------------------

#define BATCH 16
#define HH 512
#define WW 512
#define NPIX (HH * WW)            // 262144
#define NTOT (BATCH * NPIX)       // 4194304
#define NITER 128
#define SIGMA (0.97f / 3.0f)
#define TAU (0.97f / 3.0f)

typedef __attribute__((ext_vector_type(2))) float v2f;
typedef __attribute__((ext_vector_type(8))) float v8f;
typedef __attribute__((ext_vector_type(4))) unsigned v4u;
typedef __attribute__((ext_vector_type(4))) int v4i;
typedef __attribute__((ext_vector_type(8))) int v8i;

// f32 WMMA: D(16x16) = A(16x4) * B(4x16) + C.
#define WMMA4(A, B, C) \
  __builtin_amdgcn_wmma_f32_16x16x4_f32(false, (A), false, (B), (short)0, (C), false, false)

// TDM issue helper: D# groups packed per CDNA5 ISA 8.3/8.4. 2D tile of
// float2 (data_size=3): tensor 512x512, tile 512x16, dim0 stride 512.
__device__ __forceinline__ void tdm_issue(unsigned lds_addr, unsigned long long gaddr,
                                          int isStore) {
  v4u g0 = {1u,                                   // count=1, gather off
            lds_addr,                             // lds_addr [63:32]
            (unsigned)gaddr,                      // global_addr lo
            (unsigned)(gaddr >> 32) | (2u << 30)};// global_addr hi | type=2
  v8i g1 = {(int)(3u << 16),                      // data_size=3 (8B), mask=0
            (int)(512u << 16),                    // tensor_dim0 = 512 (lo16)
            (int)(512u << 16),                    // dim0 hi=0 | tensor_dim1 lo16 = 512
            (int)(512u << 16),                    // dim1 hi=0 | tile_dim0 = 512
            (int)16,                              // tile_dim1 = 16, tile_dim2 = 0
            (int)512,                             // tensor_dim0_stride = 512
            0, 0};                                // stride hi, dim1_stride = 0
  v4i g2 = {0, 0, 0, 0};
  v4i g3 = {0, 0, 0, 0};
#if defined(__clang_major__) && (__clang_major__ >= 23)
  v8i gx = {0, 0, 0, 0, 0, 0, 0, 0};
  if (isStore) __builtin_amdgcn_tensor_store_from_lds(g0, g1, g2, g3, gx, 0);
  else         __builtin_amdgcn_tensor_load_to_lds(g0, g1, g2, g3, gx, 0);
#else
  if (isStore) __builtin_amdgcn_tensor_store_from_lds(g0, g1, g2, g3, 0);
  else         __builtin_amdgcn_tensor_load_to_lds(g0, g1, g2, g3, 0);
#endif
}

// Matrix table layout (floats), per direction (fwd @0, inv @3584):
//  [0,1024)   w32r   [1024,2048) w32i     (W32[n][k], 32x32)
//  [2048,2304) w16r  [2304,2560) w16i     (W16[a][b], 16x16, symmetric)
//  [2560,3072) twr   [3072,3584) twi      (TWc[n1][k2], 16x32, centered signs)
#define MATS_PER_DIR 3584

__global__ __launch_bounds__(256) void k_mats(float* __restrict__ mats) {
  const float TWO_PI = 6.283185307179586f;
  for (int d = 0; d < 2; ++d) {
    float sgn = d ? 1.0f : -1.0f;
    float* base = mats + d * MATS_PER_DIR;
    for (int i = threadIdx.x; i < 1024; i += 256) {
      int nn = i >> 5, kk = i & 31;
      float ang = sgn * TWO_PI * (float)((nn * kk) & 31) * (1.0f / 32.0f);
      base[i] = cosf(ang);
      base[1024 + i] = sinf(ang);
    }
    for (int i = threadIdx.x; i < 256; i += 256) {
      int nn = i >> 4, kk = i & 15;
      float ang = sgn * TWO_PI * (float)((nn * kk) & 15) * (1.0f / 16.0f);
      base[2048 + i] = cosf(ang);
      base[2304 + i] = sinf(ang);
    }
    for (int i = threadIdx.x; i < 512; i += 256) {
      int n1 = i >> 5, k2 = i & 31;
      float ang = sgn * TWO_PI * (float)(n1 * k2) * (1.0f / 512.0f);
      float cs = ((n1 + k2) & 1) ? -1.0f : 1.0f;   // centered-FFT sign fold
      base[2560 + i] = cs * cosf(ang);
      base[3072 + i] = cs * sinf(ang);
    }
  }
}

// ---------------------------------------------------------------------------
// One 1D centered-DFT pass over 512-point lines, 16 lines per block.
// colPass=0: lines are image rows (contiguous; staged by TDM).
// colPass=1: lines are columns (transposed through LDS, manual loads).
// out[32*k1+k2] = sum_n1 W16[k1][n1]*TWc[n1][k2]* sum_n2 X[n1+16*n2]*W32[n2][k2]
// ---------------------------------------------------------------------------
__global__ __launch_bounds__(256) void fft512_pass(
    const float2* __restrict__ in, float2* __restrict__ out,
    const float* __restrict__ mats, int colPass) {
  __shared__ float2 tile[16][512];          // 64KB line buffer (TDM target)
  __shared__ float2 zscr[8][16][32];        // per-wave intermediate Z
  __shared__ float lw32r[1024], lw32i[1024];
  __shared__ float lw16r[256], lw16i[256];
  __shared__ float ltwr[512], ltwi[512];

  const int t = threadIdx.x;
  const int lane = t & 31;
  const int wave = t >> 5;
  const int b = blockIdx.x >> 5;            // image index
  const int base16 = (blockIdx.x & 31) << 4;// first row/col of this group
  const float2* imgIn = in + (size_t)b * NPIX;
  float2* imgOut = out + (size_t)b * NPIX;

  // Row pass: kick off the TDM tile load first (wave 0 only; TDM ignores EXEC).
  if (!colPass && wave == 0) {
    unsigned ldsA = (unsigned)(unsigned long long)(void*)&tile[0][0];
    tdm_issue(ldsA, (unsigned long long)(imgIn + (size_t)base16 * WW), 0);
  }

  // Stage DFT matrices while the DMA runs.
  for (int i = t; i < 1024; i += 256) { lw32r[i] = mats[i]; lw32i[i] = mats[1024 + i]; }
  for (int i = t; i < 256; i += 256)  { lw16r[i] = mats[2048 + i]; lw16i[i] = mats[2304 + i]; }
  for (int i = t; i < 512; i += 256)  { ltwr[i] = mats[2560 + i]; ltwi[i] = mats[3072 + i]; }

  if (!colPass) {
    if (wave == 0) __builtin_amdgcn_s_wait_tensorcnt(0);
  } else {
    for (int i = 0; i < 32; ++i) {
      int li = t + 256 * i; int r = li >> 4; int cl = li & 15;
      tile[cl][r] = imgIn[(size_t)r * WW + base16 + cl];
    }
  }
  __syncthreads();

  const int mrow = lane & 15;
  const int hiHalf = lane >> 4;             // 0: K+0/K+1, 1: K+2/K+3
  const float scale = 0.044194173824159216f; // 1/sqrt(512) (ortho per pass)

  for (int rep = 0; rep < 2; ++rep) {
    const int line = wave * 2 + rep;

    // ---- step 1: Y(16x32) = X1(16x32) * W32(32x32), twiddle -> zscr ----
    for (int tt = 0; tt < 2; ++tt) {
      v8f accRa = {0,0,0,0,0,0,0,0}, accRb = {0,0,0,0,0,0,0,0};
      v8f accIa = {0,0,0,0,0,0,0,0}, accIb = {0,0,0,0,0,0,0,0};
      const int n = mrow + 16 * tt;
#pragma unroll
      for (int k0 = 0; k0 < 32; k0 += 4) {
        const int kk = k0 + 2 * hiHalf;
        float2 e0 = tile[line][mrow + 16 * kk];
        float2 e1 = tile[line][mrow + 16 * (kk + 1)];
        v2f Ar = {e0.x, e1.x}, Ai = {e0.y, e1.y};
        v2f Br = {lw32r[kk * 32 + n], lw32r[(kk + 1) * 32 + n]};
        v2f Bi = {lw32i[kk * 32 + n], lw32i[(kk + 1) * 32 + n]};
        accRa = WMMA4(Ar, Br, accRa);
        accRb = WMMA4(Ai, Bi, accRb);     // subtracted later: Re = Ra - Rb
        accIa = WMMA4(Ar, Bi, accIa);
        accIb = WMMA4(Ai, Br, accIb);
      }
#pragma unroll
      for (int r = 0; r < 8; ++r) {
        int row = r + 8 * hiHalf;           // n1
        int col = mrow + 16 * tt;           // k2
        float yr = accRa[r] - accRb[r];
        float yi = accIa[r] + accIb[r];
        float tr = ltwr[row * 32 + col], ti = ltwi[row * 32 + col];
        float2 z; z.x = yr * tr - yi * ti; z.y = yr * ti + yi * tr;
        zscr[wave][row][col] = z;
      }
    }

    // ---- step 2: out(16x32) = W16(16x16) * Z(16x32) ----
    for (int tt = 0; tt < 2; ++tt) {
      v8f accRa = {0,0,0,0,0,0,0,0}, accRb = {0,0,0,0,0,0,0,0};
      v8f accIa = {0,0,0,0,0,0,0,0}, accIb = {0,0,0,0,0,0,0,0};
      const int n = mrow + 16 * tt;
#pragma unroll
      for (int k0 = 0; k0 < 16; k0 += 4) {
        const int kk = k0 + 2 * hiHalf;
        v2f Ar = {lw16r[mrow * 16 + kk], lw16r[mrow * 16 + kk + 1]};
        v2f Ai = {lw16i[mrow * 16 + kk], lw16i[mrow * 16 + kk + 1]};
        float2 z0 = zscr[wave][kk][n];
        float2 z1 = zscr[wave][kk + 1][n];
        v2f Br = {z0.x, z1.x}, Bi = {z0.y, z1.y};
        accRa = WMMA4(Ar, Br, accRa);
        accRb = WMMA4(Ai, Bi, accRb);
        accIa = WMMA4(Ar, Bi, accIa);
        accIb = WMMA4(Ai, Br, accIb);
      }
#pragma unroll
      for (int r = 0; r < 8; ++r) {
        int k1 = r + 8 * hiHalf;
        int k2 = mrow + 16 * tt;
        float2 o;
        o.x = (accRa[r] - accRb[r]) * scale;
        o.y = (accIa[r] + accIb[r]) * scale;
        tile[line][32 * k1 + k2] = o;
      }
    }
  }
  __syncthreads();

  if (!colPass) {
    // TDM store of the finished tile (wave 0; drained before ENDPGM).
    if (wave == 0) {
      unsigned ldsA = (unsigned)(unsigned long long)(void*)&tile[0][0];
      tdm_issue(ldsA, (unsigned long long)(imgOut + (size_t)base16 * WW), 1);
      __builtin_amdgcn_s_wait_tensorcnt(0);
    }
  } else {
    for (int i = 0; i < 32; ++i) {
      int li = t + 256 * i; int r = li >> 4; int cl = li & 15;
      imgOut[(size_t)r * WW + base16 + cl] = tile[cl][r];
    }
  }
}

// ---------------------------------------------------------------------------
// Elementwise PDHG kernels
// ---------------------------------------------------------------------------
__global__ __launch_bounds__(256) void k_init(
    const float* __restrict__ xr, const float* __restrict__ xi,
    float2* __restrict__ x, float2* __restrict__ xbar,
    float2* __restrict__ p1, float2* __restrict__ p2a, float2* __restrict__ p2b) {
  int i = blockIdx.x * 256 + threadIdx.x;
  float2 v; v.x = xr[i]; v.y = xi[i];
  x[i] = v; xbar[i] = v;
  float2 z; z.x = 0.0f; z.y = 0.0f;
  p1[i] = z; p2a[i] = z; p2b[i] = z;
}

__global__ __launch_bounds__(256) void k_p1(
    float2* __restrict__ p1, const float2* __restrict__ khat,
    const float* __restrict__ yr, const float* __restrict__ yi,
    const int* __restrict__ mask, float2* __restrict__ mp1) {
  int i = blockIdx.x * 256 + threadIdx.x;
  int pix = i & (NPIX - 1);
  float m = (float)mask[pix];
  float2 kh = khat[i];
  float2 p = p1[i];
  const float inv = 1.0f / (1.0f + SIGMA);
  p.x = (p.x + SIGMA * (m * kh.x) - SIGMA * yr[i]) * inv;
  p.y = (p.y + SIGMA * (m * kh.y) - SIGMA * yi[i]) * inv;
  p1[i] = p;
  float2 o; o.x = m * p.x; o.y = m * p.y;
  mp1[i] = o;
}

__global__ __launch_bounds__(256) void k_p2(
    const float2* __restrict__ xbar, float2* __restrict__ p2a,
    float2* __restrict__ p2b, const float* __restrict__ lam) {
  int i = blockIdx.x * 256 + threadIdx.x;
  int pix = i & (NPIX - 1);
  int base = i - pix;
  int r = pix >> 9, c = pix & 511;
  float2 xb = xbar[i];
  float2 xd = xbar[base + (((r + 1) & 511) << 9) + c];       // roll -1 along H
  float2 xrt = xbar[base + (r << 9) + ((c + 1) & 511)];      // roll -1 along W
  float l = lam[i];
  float2 qa = p2a[i], qb = p2b[i];
  qa.x = fminf(fmaxf(qa.x + SIGMA * (xd.x - xb.x), -l), l);
  qa.y = fminf(fmaxf(qa.y + SIGMA * (xd.y - xb.y), -l), l);
  qb.x = fminf(fmaxf(qb.x + SIGMA * (xrt.x - xb.x), -l), l);
  qb.y = fminf(fmaxf(qb.y + SIGMA * (xrt.y - xb.y), -l), l);
  p2a[i] = qa; p2b[i] = qb;
}

__global__ __launch_bounds__(256) void k_x(
    float2* __restrict__ x, float2* __restrict__ xbar,
    const float2* __restrict__ g, const float2* __restrict__ p2a,
    const float2* __restrict__ p2b) {
  int i = blockIdx.x * 256 + threadIdx.x;
  int pix = i & (NPIX - 1);
  int base = i - pix;
  int r = pix >> 9, c = pix & 511;
  float2 au = p2a[base + (((r + 511) & 511) << 9) + c];      // roll +1 along H
  float2 al = p2b[base + (r << 9) + ((c + 511) & 511)];      // roll +1 along W
  float2 pa = p2a[i], pb = p2b[i];
  float2 adj; adj.x = (au.x - pa.x) + (al.x - pb.x);
  adj.y = (au.y - pa.y) + (al.y - pb.y);
  float2 gv = g[i];
  float2 xv = x[i];
  float2 xn;
  xn.x = xv.x - TAU * (gv.x + adj.x);
  xn.y = xv.y - TAU * (gv.y + adj.y);
  float2 xb; xb.x = 2.0f * xn.x - xv.x; xb.y = 2.0f * xn.y - xv.y;
  x[i] = xn; xbar[i] = xb;
}

// ---------------------------------------------------------------------------
extern "C" void kernel_launch(void* const* d_in, const int* in_sizes, int n_in,
                              void* d_out, int out_size, void* d_ws, size_t ws_size,
                              hipStream_t stream) {
  (void)in_sizes; (void)n_in; (void)out_size; (void)ws_size;
  const float* in_r = (const float*)d_in[0];
  const float* in_i = (const float*)d_in[1];
  const float* kd_r = (const float*)d_in[2];
  const float* kd_i = (const float*)d_in[3];
  const float* lam  = (const float*)d_in[4];
  const int*   mask = (const int*)d_in[5];

  float2* x = (float2*)d_out;  // output layout (...,2) == interleaved complex

  char* ws = (char*)d_ws;
  const size_t PLANE = (size_t)NTOT * sizeof(float2);
  float2* xbar = (float2*)(ws + 0 * PLANE);
  float2* p1   = (float2*)(ws + 1 * PLANE);
  float2* p2a  = (float2*)(ws + 2 * PLANE);
  float2* p2b  = (float2*)(ws + 3 * PLANE);
  float2* tmpA = (float2*)(ws + 4 * PLANE);
  float2* tmpK = (float2*)(ws + 5 * PLANE);
  float*  mats = (float*)(ws + 6 * PLANE);

  const int EB = NTOT / 256;       // elementwise blocks
  const int FB = 512;              // fft blocks: 16 imgs * 32 groups of 16 lines

  k_mats<<<1, 256, 0, stream>>>(mats);
  k_init<<<EB, 256, 0, stream>>>(in_r, in_i, x, xbar, p1, p2a, p2b);

  for (int it = 0; it < NITER; ++it) {
    // khat = fft2c(xbar): centered-DFT rows then cols (signs in twiddles)
    fft512_pass<<<FB, 256, 0, stream>>>(xbar, tmpA, mats, 0);
    fft512_pass<<<FB, 256, 0, stream>>>(tmpA, tmpK, mats, 1);
    // p1 prox; tmpA = m * p1
    k_p1<<<EB, 256, 0, stream>>>(p1, tmpK, kd_r, kd_i, mask, tmpA);
    // g = ifft2c(tmpA)
    fft512_pass<<<FB, 256, 0, stream>>>(tmpA, tmpK, mats + MATS_PER_DIR, 0);
    fft512_pass<<<FB, 256, 0, stream>>>(tmpK, tmpA, mats + MATS_PER_DIR, 1);
    // TV dual prox (reads old xbar), then primal step + extrapolation
    k_p2<<<EB, 256, 0, stream>>>(xbar, p2a, p2b, lam);
    k_x<<<EB, 256, 0, stream>>>(x, xbar, tmpA, p2a, p2b);
  }
}